// PerformerAttention_25812753449199
// MI455X (gfx1250) — compile-verified
//
#include <hip/hip_runtime.h>
#include <hip/hip_bf16.h>

// ---------------- problem constants ----------------
#define BATCH   4
#define SEQ     2048
#define EMB     512
#define NHEAD   8
#define HEAD    64
#define FEAT    512              // random-feature dim m
#define NROWS   (BATCH*NHEAD*SEQ)   // 65536 rows over (b,h,n)

#define LDSA 33                  // 32 + 1 pad (odd -> conflict-free row reads)
#define LDSB 65                  // 64 + 1 pad

typedef __attribute__((ext_vector_type(2))) float v2f;
typedef __attribute__((ext_vector_type(8))) float v8f;

// ---------------- wave-level WMMA f32 16x16x4 over one K-tile of 32 ----------
// All fragments of the K-tile are prefetched into VGPR arrays first so the
// LDS loads batch under a single dscnt wait, then 16 WMMAs issue back-to-back.
// As: [64][LDSA] (rows x k), Bs: [32][LDSB] (k x cols). Wave computes a 16x32
// strip: rows 16*wm.., cols cn0..cn0+31 (two 16x16 accumulators).
__device__ __forceinline__ void wave_mma_ktile(const float* __restrict__ As,
                                               const float* __restrict__ Bs,
                                               v8f& acc0, v8f& acc1,
                                               int wm, int cn0, int lane) {
  const int half = lane >> 4;      // 0: K pair 0/1, 1: K pair 2/3
  const int lm   = lane & 15;
  const int arow = wm * 16 + lm;
  v2f af[8], bf0[8], bf1[8];
#pragma unroll
  for (int t = 0; t < 8; ++t) {
    const int ka = t * 4 + 2 * half;
    af[t].x  = As[arow * LDSA + ka];
    af[t].y  = As[arow * LDSA + ka + 1];
    bf0[t].x = Bs[ka * LDSB + cn0 + lm];
    bf0[t].y = Bs[(ka + 1) * LDSB + cn0 + lm];
    bf1[t].x = Bs[ka * LDSB + cn0 + 16 + lm];
    bf1[t].y = Bs[(ka + 1) * LDSB + cn0 + 16 + lm];
  }
#pragma unroll
  for (int t = 0; t < 8; ++t) {
    acc0 = __builtin_amdgcn_wmma_f32_16x16x4_f32(false, af[t], false, bf0[t], (short)0, acc0, false, false);
    acc1 = __builtin_amdgcn_wmma_f32_16x16x4_f32(false, af[t], false, bf1[t], (short)0, acc1, false, false);
  }
}

// ---------------- GEMM 1/6: C(MxN) = A(MxK) @ B(KxN) + bias, row-major -------
// Double-buffered: next K-tile's global loads are issued right after the
// barrier and complete under the WMMA chain.
__global__ __launch_bounds__(256) void gemm_nn_bias(
    const float* __restrict__ A, const float* __restrict__ B,
    const float* __restrict__ bias, float* __restrict__ C,
    int K, int lda, int ldb, int ldc) {
  __shared__ float As[64 * LDSA];
  __shared__ float Bs[32 * LDSB];
  const int tid = threadIdx.x, lane = tid & 31, wave = tid >> 5;
  const int wm = wave & 3, wc = wave >> 2;
  const int rowBase = blockIdx.y * 64, colBase = blockIdx.x * 64;
  const int ar0 = tid >> 5, ac = tid & 31;   // A stage: rows ar0+8i, col ac
  const int br0 = tid >> 6, bc = tid & 63;   // B stage: rows br0+4i, col bc
  float ra[8], rb[8];
  v8f acc0 = {}; v8f acc1 = {};
#pragma unroll
  for (int i = 0; i < 8; ++i) {
    ra[i] = A[(size_t)(rowBase + ar0 + i * 8) * lda + ac];
    rb[i] = B[(size_t)(br0 + i * 4) * ldb + colBase + bc];
  }
  for (int k0 = 0; k0 < K; k0 += 32) {
#pragma unroll
    for (int i = 0; i < 8; ++i) {
      As[(ar0 + i * 8) * LDSA + ac] = ra[i];
      Bs[(br0 + i * 4) * LDSB + bc] = rb[i];
    }
    __syncthreads();
    const int k1 = k0 + 32;
    if (k1 < K) {
#pragma unroll
      for (int i = 0; i < 8; ++i) {
        ra[i] = A[(size_t)(rowBase + ar0 + i * 8) * lda + k1 + ac];
        rb[i] = B[(size_t)(k1 + br0 + i * 4) * ldb + colBase + bc];
      }
    }
    wave_mma_ktile(As, Bs, acc0, acc1, wm, wc * 32, lane);
    __syncthreads();
  }
  const int half = lane >> 4, lm = lane & 15;
  const int col0 = colBase + wc * 32 + lm;
#pragma unroll
  for (int r = 0; r < 8; ++r) {
    int row = rowBase + wm * 16 + half * 8 + r;
    C[(size_t)row * ldc + col0]      = acc0[r] + bias[col0];
    C[(size_t)row * ldc + col0 + 16] = acc1[r] + bias[col0 + 16];
  }
}

// ---------------- GEMM 2: data_dash = dn * Qh @ proj^T  (per b,h) ------------
// A = head view of Q/K raw (b,n,EMB), lda=EMB, K=HEAD. B[k][n] = proj[n*HEAD+k].
__global__ __launch_bounds__(256) void gemm_proj_kernel(
    const float* __restrict__ QK, const float* __restrict__ proj,
    float* __restrict__ outP, float dn) {
  const int bh = blockIdx.z, b = bh >> 3, h = bh & 7;
  const float* A = QK + (size_t)b * SEQ * EMB + h * HEAD;
  float* C = outP + (size_t)bh * SEQ * FEAT;
  __shared__ float As[64 * LDSA];
  __shared__ float Bs[32 * LDSB];
  const int tid = threadIdx.x, lane = tid & 31, wave = tid >> 5;
  const int wm = wave & 3, wc = wave >> 2;
  const int rowBase = blockIdx.y * 64, colBase = blockIdx.x * 64;
  const int ar0 = tid >> 5, ac = tid & 31;
  const int bn0 = tid >> 5, bk = tid & 31;   // transposed B staging
  float ra[8], rb[8];
  v8f acc0 = {}; v8f acc1 = {};
#pragma unroll
  for (int i = 0; i < 8; ++i) {
    ra[i] = A[(size_t)(rowBase + ar0 + i * 8) * EMB + ac];
    rb[i] = proj[(size_t)(colBase + bn0 + i * 8) * HEAD + bk];
  }
  for (int k0 = 0; k0 < HEAD; k0 += 32) {
#pragma unroll
    for (int i = 0; i < 8; ++i) {
      As[(ar0 + i * 8) * LDSA + ac] = ra[i];
      Bs[bk * LDSB + bn0 + i * 8]   = rb[i];
    }
    __syncthreads();
    const int k1 = k0 + 32;
    if (k1 < HEAD) {
#pragma unroll
      for (int i = 0; i < 8; ++i) {
        ra[i] = A[(size_t)(rowBase + ar0 + i * 8) * EMB + k1 + ac];
        rb[i] = proj[(size_t)(colBase + bn0 + i * 8) * HEAD + k1 + bk];
      }
    }
    wave_mma_ktile(As, Bs, acc0, acc1, wm, wc * 32, lane);
    __syncthreads();
  }
  const int half = lane >> 4, lm = lane & 15;
  const int col0 = colBase + wc * 32 + lm;
#pragma unroll
  for (int r = 0; r < 8; ++r) {
    int row = rowBase + wm * 16 + half * 8 + r;
    C[(size_t)row * FEAT + col0]      = dn * acc0[r];
    C[(size_t)row * FEAT + col0 + 16] = dn * acc1[r];
  }
}

// ---------------- GEMM 4: context = KP^T @ Vh  (per b,h): (FEATx64) ----------
__global__ __launch_bounds__(256) void gemm_ctx_kernel(
    const float* __restrict__ KP, const float* __restrict__ V,
    float* __restrict__ ctx) {
  const int bh = blockIdx.z, b = bh >> 3, h = bh & 7;
  const float* Ab = KP + (size_t)bh * SEQ * FEAT;          // A[m][k] = Ab[k*FEAT+m]
  const float* Bb = V + (size_t)b * SEQ * EMB + h * HEAD;  // B[k][n] = Bb[k*EMB+n]
  float* C = ctx + (size_t)bh * FEAT * HEAD;               // ldc=HEAD
  __shared__ float As[64 * LDSA];
  __shared__ float Bs[32 * LDSB];
  const int tid = threadIdx.x, lane = tid & 31, wave = tid >> 5;
  const int wm = wave & 3, wc = wave >> 2;
  const int rowBase = blockIdx.y * 64;                     // feature dim m
  const int ak0 = tid >> 6, am = tid & 63;                 // transposed A staging
  const int br0 = tid >> 6, bc = tid & 63;
  float ra[8], rb[8];
  v8f acc0 = {}; v8f acc1 = {};
#pragma unroll
  for (int i = 0; i < 8; ++i) {
    ra[i] = Ab[(size_t)(ak0 + i * 4) * FEAT + rowBase + am];
    rb[i] = Bb[(size_t)(br0 + i * 4) * EMB + bc];
  }
  for (int k0 = 0; k0 < SEQ; k0 += 32) {
#pragma unroll
    for (int i = 0; i < 8; ++i) {
      As[am * LDSA + ak0 + i * 4]   = ra[i];
      Bs[(br0 + i * 4) * LDSB + bc] = rb[i];
    }
    __syncthreads();
    const int k1 = k0 + 32;
    if (k1 < SEQ) {
#pragma unroll
      for (int i = 0; i < 8; ++i) {
        ra[i] = Ab[(size_t)(k1 + ak0 + i * 4) * FEAT + rowBase + am];
        rb[i] = Bb[(size_t)(k1 + br0 + i * 4) * EMB + bc];
      }
    }
    wave_mma_ktile(As, Bs, acc0, acc1, wm, wc * 32, lane);
    __syncthreads();
  }
  const int half = lane >> 4, lm = lane & 15;
  const int col0 = wc * 32 + lm;
#pragma unroll
  for (int r = 0; r < 8; ++r) {
    int row = rowBase + wm * 16 + half * 8 + r;
    C[(size_t)row * HEAD + col0]      = acc0[r];
    C[(size_t)row * HEAD + col0 + 16] = acc1[r];
  }
}

// ---------------- GEMM 5: out = (QP @ ctx) * dinv, merged into (b,n,h*d) -----
__global__ __launch_bounds__(256) void gemm_out_kernel(
    const float* __restrict__ QP, const float* __restrict__ ctx,
    const float* __restrict__ dinv, float* __restrict__ attn) {
  const int bh = blockIdx.z, b = bh >> 3, h = bh & 7;
  const float* A = QP + (size_t)bh * SEQ * FEAT;
  const float* B = ctx + (size_t)bh * FEAT * HEAD;         // ldb=HEAD
  float* C = attn + (size_t)b * SEQ * EMB + h * HEAD;      // ldc=EMB
  __shared__ float As[64 * LDSA];
  __shared__ float Bs[32 * LDSB];
  const int tid = threadIdx.x, lane = tid & 31, wave = tid >> 5;
  const int wm = wave & 3, wc = wave >> 2;
  const int rowBase = blockIdx.y * 64;
  const int ar0 = tid >> 5, ac = tid & 31;
  const int br0 = tid >> 6, bc = tid & 63;
  float ra[8], rb[8];
  v8f acc0 = {}; v8f acc1 = {};
#pragma unroll
  for (int i = 0; i < 8; ++i) {
    ra[i] = A[(size_t)(rowBase + ar0 + i * 8) * FEAT + ac];
    rb[i] = B[(size_t)(br0 + i * 4) * HEAD + bc];
  }
  for (int k0 = 0; k0 < FEAT; k0 += 32) {
#pragma unroll
    for (int i = 0; i < 8; ++i) {
      As[(ar0 + i * 8) * LDSA + ac] = ra[i];
      Bs[(br0 + i * 4) * LDSB + bc] = rb[i];
    }
    __syncthreads();
    const int k1 = k0 + 32;
    if (k1 < FEAT) {
#pragma unroll
      for (int i = 0; i < 8; ++i) {
        ra[i] = A[(size_t)(rowBase + ar0 + i * 8) * FEAT + k1 + ac];
        rb[i] = B[(size_t)(k1 + br0 + i * 4) * HEAD + bc];
      }
    }
    wave_mma_ktile(As, Bs, acc0, acc1, wm, wc * 32, lane);
    __syncthreads();
  }
  const int half = lane >> 4, lm = lane & 15;
  const int col0 = wc * 32 + lm;
#pragma unroll
  for (int r = 0; r < 8; ++r) {
    int row = rowBase + wm * 16 + half * 8 + r;
    float s = dinv[(size_t)bh * SEQ + row];
    C[(size_t)row * EMB + col0]      = acc0[r] * s;
    C[(size_t)row * EMB + col0 + 16] = acc1[r] * s;
  }
}

// ---------------- reductions / elementwise ----------------------------------
// diag[r] = coef * sum_d q[b,n,h*64+d]^2   (one wave32 per row)
__global__ __launch_bounds__(256) void diag_kernel(const float* __restrict__ QK,
                                                   float* __restrict__ diag,
                                                   float coef) {
  const int r = blockIdx.x * 8 + (threadIdx.x >> 5);
  const int lane = threadIdx.x & 31;
  const int b = r >> 14, h = (r >> 11) & 7, n = r & 2047;
  const float* p = QK + (size_t)b * SEQ * EMB + (size_t)n * EMB + h * HEAD;
  float v0 = p[lane], v1 = p[lane + 32];
  float s = v0 * v0 + v1 * v1;
#pragma unroll
  for (int m = 16; m >= 1; m >>= 1) s += __shfl_xor(s, m, 32);
  if (lane == 0) diag[r] = coef * s;
}

// rowmax[r] = max_m QP[r][m]   (one wave32 per row)
__global__ __launch_bounds__(256) void rowmax_kernel(const float* __restrict__ P,
                                                     float* __restrict__ rmax) {
  const int r = blockIdx.x * 8 + (threadIdx.x >> 5);
  const int lane = threadIdx.x & 31;
  const float* p = P + (size_t)r * FEAT;
  float m = -3.0e38f;
#pragma unroll
  for (int i = 0; i < 16; ++i) m = fmaxf(m, p[lane + i * 32]);
#pragma unroll
  for (int s = 16; s >= 1; s >>= 1) m = fmaxf(m, __shfl_xor(m, s, 32));
  if (lane == 0) rmax[r] = m;
}

// hmax[bh] = max over SEQ*FEAT of KP head slab
__global__ __launch_bounds__(256) void headmax_kernel(const float* __restrict__ P,
                                                      float* __restrict__ hmax) {
  __shared__ float red[8];
  const int bh = blockIdx.x, tid = threadIdx.x;
  const float* p = P + (size_t)bh * SEQ * FEAT;
  float m = -3.0e38f;
  for (int i = tid; i < SEQ * FEAT; i += 256) m = fmaxf(m, p[i]);
#pragma unroll
  for (int s = 16; s >= 1; s >>= 1) m = fmaxf(m, __shfl_xor(m, s, 32));
  if ((tid & 31) == 0) red[tid >> 5] = m;
  __syncthreads();
  if (tid == 0) {
    float mm = red[0];
#pragma unroll
    for (int i = 1; i < 8; ++i) mm = fmaxf(mm, red[i]);
    hmax[bh] = mm;
  }
}

// P[r][m] = ratio * (exp(P - diag[r] - mx[r>>mshift]) + 1e-4), float4 vectorized
__global__ __launch_bounds__(256) void phi_finalize(float* __restrict__ P,
                                                    const float* __restrict__ diag,
                                                    const float* __restrict__ mx,
                                                    int mshift, float ratio) {
  const size_t base = ((size_t)blockIdx.x * 256 + threadIdx.x) * 4;
  const int r = (int)(base >> 9);                    // /FEAT
  const float sub = diag[r] + mx[r >> mshift];
  float4 v = *reinterpret_cast<const float4*>(P + base);
  v.x = ratio * (__expf(v.x - sub) + 1e-4f);
  v.y = ratio * (__expf(v.y - sub) + 1e-4f);
  v.z = ratio * (__expf(v.z - sub) + 1e-4f);
  v.w = ratio * (__expf(v.w - sub) + 1e-4f);
  *reinterpret_cast<float4*>(P + base) = v;
}

// ksum[bh][m] = sum_n KP[bh][n][m]
__global__ __launch_bounds__(256) void kcumsum_kernel(const float* __restrict__ KP,
                                                      float* __restrict__ ksum) {
  const int t = blockIdx.x * 256 + threadIdx.x;      // 16384 threads
  const int bh = t >> 9, m = t & 511;
  const float* p = KP + (size_t)bh * SEQ * FEAT + m;
  float s = 0.f;
  for (int n = 0; n < SEQ; ++n) s += p[(size_t)n * FEAT];
  ksum[t] = s;
}

// dinv[r] = 1 / (QP[r][:] . ksum[bh][:])   (one wave32 per row)
__global__ __launch_bounds__(256) void dinv_kernel(const float* __restrict__ QP,
                                                   const float* __restrict__ ksum,
                                                   float* __restrict__ dinv) {
  const int r = blockIdx.x * 8 + (threadIdx.x >> 5);
  const int lane = threadIdx.x & 31;
  const int bh = r >> 11;
  const float* p = QP + (size_t)r * FEAT;
  const float* ks = ksum + (size_t)bh * FEAT;
  float s = 0.f;
#pragma unroll
  for (int i = 0; i < 16; ++i) {
    int m = lane + i * 32;
    s += p[m] * ks[m];
  }
#pragma unroll
  for (int m = 16; m >= 1; m >>= 1) s += __shfl_xor(s, m, 32);
  if (lane == 0) dinv[r] = 1.f / s;
}

// ---------------- host-side orchestration -----------------------------------
extern "C" void kernel_launch(void* const* d_in, const int* in_sizes, int n_in,
                              void* d_out, int out_size, void* d_ws, size_t ws_size,
                              hipStream_t stream) {
  (void)in_sizes; (void)n_in; (void)out_size; (void)ws_size;
  const float* x    = (const float*)d_in[0];
  const float* Wq   = (const float*)d_in[1];
  const float* bq   = (const float*)d_in[2];
  const float* Wk   = (const float*)d_in[3];
  const float* bk   = (const float*)d_in[4];
  const float* Wv   = (const float*)d_in[5];
  const float* bv   = (const float*)d_in[6];
  const float* Wo   = (const float*)d_in[7];
  const float* bo   = (const float*)d_in[8];
  const float* proj = (const float*)d_in[9];
  float* out = (float*)d_out;

  // workspace layout (floats)
  float* ws = (float*)d_ws;
  const size_t szQKV = (size_t)BATCH * SEQ * EMB;          // 4,194,304
  const size_t szP   = (size_t)BATCH * NHEAD * SEQ * FEAT; // 33,554,432
  float* Qb     = ws;                 // raw Q (b,n,E); reused later as attn_out
  float* Kb     = Qb + szQKV;
  float* Vb     = Kb + szQKV;
  float* QP     = Vb + szQKV;         // (bh,n,FEAT)
  float* KP     = QP + szP;
  float* ctx    = KP + szP;           // (bh,FEAT,HEAD)
  float* ksum   = ctx + (size_t)BATCH * NHEAD * FEAT * HEAD;
  float* diag_q = ksum + (size_t)BATCH * NHEAD * FEAT;
  float* diag_k = diag_q + NROWS;
  float* rmax_q = diag_k + NROWS;
  float* hmax_k = rmax_q + NROWS;
  float* dinv   = hmax_k + 32;
  float* attn   = Qb;                 // overlay: Qb dead once QP+diag_q done

  const float dn    = 0.3535533905932738f;   // 64^-0.25
  const float coef  = 0.0625f;               // 0.5*dn*dn
  const float ratio = 0.04419417382415922f;  // 512^-0.5

  dim3 blk(256);
  // 1) Q/K/V = x @ W + b     (8192 x 512 x 512)
  dim3 gXW(EMB / 64, (BATCH * SEQ) / 64, 1);
  gemm_nn_bias<<<gXW, blk, 0, stream>>>(x, Wq, bq, Qb, EMB, EMB, EMB, EMB);
  gemm_nn_bias<<<gXW, blk, 0, stream>>>(x, Wk, bk, Kb, EMB, EMB, EMB, EMB);
  gemm_nn_bias<<<gXW, blk, 0, stream>>>(x, Wv, bv, Vb, EMB, EMB, EMB, EMB);

  // 2) raw data_dash for Q and K  (per-head 2048 x 512, K=64)
  dim3 gP(FEAT / 64, SEQ / 64, BATCH * NHEAD);
  gemm_proj_kernel<<<gP, blk, 0, stream>>>(Qb, proj, QP, dn);
  gemm_proj_kernel<<<gP, blk, 0, stream>>>(Kb, proj, KP, dn);

  // 3) diag, maxima, exp-finalize
  diag_kernel<<<NROWS / 8, blk, 0, stream>>>(Qb, diag_q, coef);
  diag_kernel<<<NROWS / 8, blk, 0, stream>>>(Kb, diag_k, coef);
  rowmax_kernel<<<NROWS / 8, blk, 0, stream>>>(QP, rmax_q);
  headmax_kernel<<<BATCH * NHEAD, blk, 0, stream>>>(KP, hmax_k);
  const int nPhiBlocks = (int)(szP / (256 * 4));
  phi_finalize<<<nPhiBlocks, blk, 0, stream>>>(QP, diag_q, rmax_q, 0, ratio);
  phi_finalize<<<nPhiBlocks, blk, 0, stream>>>(KP, diag_k, hmax_k, 11, ratio);

  // 4) k_cumsum, context = KP^T @ V
  kcumsum_kernel<<<(BATCH * NHEAD * FEAT) / 256, blk, 0, stream>>>(KP, ksum);
  dim3 gC(1, FEAT / 64, BATCH * NHEAD);
  gemm_ctx_kernel<<<gC, blk, 0, stream>>>(KP, Vb, ctx);

  // 5) d_inv, out = (QP @ ctx) * d_inv  -> merged-head layout (b,n,h*d)
  dinv_kernel<<<NROWS / 8, blk, 0, stream>>>(QP, ksum, dinv);
  dim3 gO(1, SEQ / 64, BATCH * NHEAD);
  gemm_out_kernel<<<gO, blk, 0, stream>>>(QP, ctx, dinv, attn);

  // 6) final projection: out = attn @ Wo + bo
  gemm_nn_bias<<<gXW, blk, 0, stream>>>(attn, Wo, bo, out, EMB, EMB, EMB, EMB);
}